// AugmentableSVSAlgorithm_88587995447542
// MI455X (gfx1250) — compile-verified
//
#include <hip/hip_runtime.h>
#include <stdint.h>

// Problem constants (fixed by the reference module)
#define HSZ 128
#define WSZ 160
#define HWSZ (HSZ * WSZ)

// Tiling: each block produces a 32x4 output tile, computes a 34x6 halo region.
#define TILE_W 32
#define TILE_H 4
#define CW (TILE_W + 2)   // 34
#define CH (TILE_H + 2)   // 6
#define CN (CW * CH)      // 204 active compute threads
#define NTHR 224          // 7 wave32s
#define UB 8              // frames per barrier batch (2000 % 8 == 0)
#define TILE_PIX (TILE_W * TILE_H)             // 128
#define OUTS_PER_BATCH (UB * TILE_PIX)         // 1024 conv outputs per batch
#define KKMAX ((OUTS_PER_BATCH + NTHR - 1) / NTHR)  // 5

// Builtin signature (from hipcc diagnostic): (int AS1*, int AS3*, imm offset, imm cpol)
typedef __attribute__((address_space(1))) int g_as1_i32;
typedef __attribute__((address_space(3))) int l_as3_i32;

// Async global -> LDS copy (CDNA5 GLOBAL_LOAD_ASYNC_TO_LDS_B32, tracked by ASYNCcnt).
__device__ __forceinline__ void copy_g2l_async(const float* g, float* l) {
#if __has_builtin(__builtin_amdgcn_global_load_async_to_lds_b32)
  __builtin_amdgcn_global_load_async_to_lds_b32(
      (g_as1_i32*)(uintptr_t)g, (l_as3_i32*)l, /*offset=*/0, /*cpol=*/0);
#else
  *l = __builtin_nontemporal_load(g);
#endif
}

__device__ __forceinline__ void async_wait0() {
#if __has_builtin(__builtin_amdgcn_s_wait_asynccnt)
  __builtin_amdgcn_s_wait_asynccnt(0);
#else
  asm volatile("s_wait_asynccnt 0" ::: "memory");
#endif
}

__device__ __forceinline__ float sigmoidf_fast(float z) {
  return 1.0f / (1.0f + __expf(-z));
}

__global__ __launch_bounds__(NTHR) void svs_fused_kernel(
    const float* __restrict__ x,       // [T,1,H,W]
    const float* __restrict__ params,  // [4] = d_close, d_open, d_hot, h_coeff
    const float* __restrict__ ht0,     // [H,W]
    const float* __restrict__ lt0,     // [H,W]
    const float* __restrict__ kern,    // [3,3]
    float* __restrict__ out,           // [T,1,H,W]
    int T)
{
  __shared__ float xt2[2][UB][CN];       // double-buffered batch of x tiles
  __shared__ float hot2f[UB * CH * CW];  // batch of hot tiles (flattened, with halo)

  const int tid = threadIdx.x;

  // Uniform scalars (compiler scalarizes these uniform loads).
  const float d_close = params[0];
  const float d_open  = params[1];
  const float d_hot   = params[2];
  const float h_coeff = params[3];
  float k[9];
#pragma unroll
  for (int i = 0; i < 9; ++i) k[i] = kern[i];

  const int  cx      = tid % CW;
  const int  cy      = tid / CW;
  const bool compute = tid < CN;
  const int  gx      = (int)blockIdx.x * TILE_W + cx - 1;
  const int  gy      = (int)blockIdx.y * TILE_H + cy - 1;
  const bool inb     = compute && ((unsigned)gx < (unsigned)WSZ) && ((unsigned)gy < (unsigned)HSZ);
  const int  goff    = gy * WSZ + gx;
  const int  soff    = cy * CW + cx;     // this thread's slot within one hot tile

  const int ox0 = (int)blockIdx.x * TILE_W;
  const int oy0 = (int)blockIdx.y * TILE_H;

  // ---- Hoisted conv assignments: loop-invariant per thread --------------------
  // j = tid + kk*NTHR enumerates the UB*128 conv outputs of a batch.
  int nconv = 0;
  int c_lds[KKMAX];  // offset of top-left 3x3 tap inside hot2f
  int c_out[KKMAX];  // u*HWSZ + global pixel offset within a frame
  int c_u[KKMAX];    // frame index within batch (for the tail path)
#pragma unroll
  for (int kk = 0; kk < KKMAX; ++kk) {
    const int j = tid + kk * NTHR;
    if (j < OUTS_PER_BATCH) {
      const int u  = j >> 7;        // / TILE_PIX
      const int p  = j & (TILE_PIX - 1);
      const int oy = p >> 5;        // 0..TILE_H-1  (output row within tile)
      const int ox = p & 31;        // 0..TILE_W-1
      c_lds[kk] = u * (CH * CW) + oy * CW + ox;   // top-left tap [oy][ox]
      c_out[kk] = u * HWSZ + (oy0 + oy) * WSZ + (ox0 + ox);
      c_u[kk]   = u;
      ++nconv;                      // wave-uniform: waves 0-3 -> 5, waves 4-6 -> 4
    }
  }

  // Per-pixel scan state (halo pixels recomputed redundantly; exact, same x).
  float ht = 0.0f, lt = 0.0f;
  if (inb) { ht = ht0[goff]; lt = lt0[goff]; }

  // Out-of-image halo cells: where_zero = 1 - hot is zero-padded, so pin hot = 1.
  if (compute && !inb) {
#pragma unroll
    for (int u = 0; u < UB; ++u) hot2f[u * (CH * CW) + soff] = 1.0f;
  }

  // Prologue: stage the first batch into buffer 0.
#pragma unroll
  for (int u = 0; u < UB; ++u)
    if (inb && u < T) copy_g2l_async(x + (size_t)u * HWSZ + goff, &xt2[0][u][tid]);
  async_wait0();
  __syncthreads();

  int buf = 0;
  // ---- Main loop: full batches only, no per-frame guards ----------------------
  for (int f = 0; f + UB <= T; f += UB) {
    const int nxt = buf ^ 1;

    // Prefetch the next batch (hidden behind this batch's compute).
    if (inb && (f + UB < T)) {
      const float* xn = x + (size_t)(f + UB) * HWSZ + goff;
      if (f + 2 * UB <= T) {        // common case: next batch is full
#pragma unroll
        for (int u = 0; u < UB; ++u)
          copy_g2l_async(xn + (size_t)u * HWSZ, &xt2[nxt][u][tid]);
      } else {                      // boundary: partial next batch (runs <= once)
#pragma unroll
        for (int u = 0; u < UB; ++u)
          if (f + UB + u < T)
            copy_g2l_async(xn + (size_t)u * HWSZ, &xt2[nxt][u][tid]);
      }
    }

    // Advance the scan through UB frames; no neighbor exchange needed.
    if (inb) {
#pragma unroll
      for (int u = 0; u < UB; ++u) {
        const float xt = xt2[buf][u][tid];
        const float th = sigmoidf_fast((xt - ht - d_hot) * h_coeff);
        const float tl = sigmoidf_fast((lt - xt - d_hot) * h_coeff);
        hot2f[u * (CH * CW) + soff] = th + tl;
        ht = (xt - ht > 0.0f) ? (ht + d_open) : (ht - d_close);
        lt = (lt - xt > 0.0f) ? (lt - d_open) : (lt + d_close);
      }
    }
    __syncthreads();  // all UB hot tiles written

    // 3x3 conv over all UB frames, work spread across ALL threads.
    float* ob = out + (size_t)f * HWSZ;
#pragma unroll
    for (int kk = 0; kk < KKMAX; ++kk) {
      if (kk < nconv) {
        const int base = c_lds[kk];
        float s = 0.0f;
#pragma unroll
        for (int i = 0; i < 3; ++i)
#pragma unroll
          for (int jj = 0; jj < 3; ++jj)
            s = fmaf(k[i * 3 + jj], 1.0f - hot2f[base + i * CW + jj], s);
        __builtin_nontemporal_store(fmaxf(1.0f - s, 0.0f), ob + c_out[kk]);
      }
    }

    async_wait0();   // next batch landed in LDS (this wave's copies)
    __syncthreads(); // ... and everyone else's; hot tiles consumed before overwrite
    buf = nxt;
  }

  // ---- Tail: T % UB frames (not taken when T is a multiple of UB) -------------
  const int Tfull = (T / UB) * UB;
  const int rem   = T - Tfull;
  if (rem > 0) {
    if (inb) {
      for (int u = 0; u < rem; ++u) {
        const float xt = xt2[buf][u][tid];
        const float th = sigmoidf_fast((xt - ht - d_hot) * h_coeff);
        const float tl = sigmoidf_fast((lt - xt - d_hot) * h_coeff);
        hot2f[u * (CH * CW) + soff] = th + tl;
        ht = (xt - ht > 0.0f) ? (ht + d_open) : (ht - d_close);
        lt = (lt - xt > 0.0f) ? (lt - d_open) : (lt + d_close);
      }
    }
    __syncthreads();
    float* ob = out + (size_t)Tfull * HWSZ;
#pragma unroll
    for (int kk = 0; kk < KKMAX; ++kk) {
      if (kk < nconv && c_u[kk] < rem) {
        const int base = c_lds[kk];
        float s = 0.0f;
#pragma unroll
        for (int i = 0; i < 3; ++i)
#pragma unroll
          for (int jj = 0; jj < 3; ++jj)
            s = fmaf(k[i * 3 + jj], 1.0f - hot2f[base + i * CW + jj], s);
        __builtin_nontemporal_store(fmaxf(1.0f - s, 0.0f), ob + c_out[kk]);
      }
    }
  }
}

extern "C" void kernel_launch(void* const* d_in, const int* in_sizes, int n_in,
                              void* d_out, int out_size, void* d_ws, size_t ws_size,
                              hipStream_t stream) {
  (void)n_in; (void)out_size; (void)d_ws; (void)ws_size;

  const float* x      = (const float*)d_in[0];  // [T,1,H,W]
  const float* params = (const float*)d_in[1];  // [4]
  const float* ht0p   = (const float*)d_in[2];  // [H,W]
  const float* lt0p   = (const float*)d_in[3];  // [H,W]
  const float* kern   = (const float*)d_in[4];  // [3,3]
  float* out          = (float*)d_out;

  const int T = in_sizes[0] / HWSZ;

  dim3 grid(WSZ / TILE_W, HSZ / TILE_H);  // 5 x 32 = 160 blocks
  svs_fused_kernel<<<grid, NTHR, 0, stream>>>(x, params, ht0p, lt0p, kern, out, T);
}